// SpearmanLoss_41875931136386
// MI455X (gfx1250) — compile-verified
//
#include <hip/hip_runtime.h>
#include <hip/hip_bf16.h>
#include <math.h>

typedef __attribute__((ext_vector_type(2))) float v2f;
typedef __attribute__((ext_vector_type(8))) float v8f;

#define N 2048
#define THREADS 256

// ---------------------------------------------------------------------------
// Kernel 1: per-row soft-rank (Blondel et al. L2 soft rank, strength = 1.0)
//   rank = s - iso_dec(s - w), un-permuted, then centered by its row mean.
// One workgroup per row; 16 rows total (8 pred + 8 target).
// Output: M[16][2048] centered ranks (rows 0-7 pred, 8-15 target).
// ---------------------------------------------------------------------------
__global__ __launch_bounds__(THREADS)
void softrank_center_kernel(const float* __restrict__ pred,
                            const float* __restrict__ target,
                            float* __restrict__ M)
{
    __shared__ float s_key[N];    // sorted values s (later: ret_sorted)
    __shared__ int   s_idx[N];    // permutation (original positions)
    __shared__ float s_z[N];      // z = -(s - w); PAVA solves iso_inc on z
    __shared__ float s_bsum[N];   // PAVA block sums
    __shared__ int   s_bcnt[N];   // PAVA block counts
    __shared__ float s_red[THREADS];
    __shared__ float s_mean;

    const int tid = threadIdx.x;
    const int row = blockIdx.x;                       // 0..15
    const float* x = (row < 8) ? (pred + row * N) : (target + (row - 8) * N);

    // load (theta = x / strength, strength == 1)
    for (int i = tid; i < N; i += THREADS) { s_key[i] = x[i]; s_idx[i] = i; }
    __syncthreads();

    // Bitonic sort, descending, key+index
    for (int k = 2; k <= N; k <<= 1) {
        for (int j = k >> 1; j > 0; j >>= 1) {
            for (int i = tid; i < N; i += THREADS) {
                int ixj = i ^ j;
                if (ixj > i) {
                    bool desc = ((i & k) == 0);       // overall descending
                    float a = s_key[i], b = s_key[ixj];
                    bool sw = desc ? (a < b) : (a > b);
                    if (sw) {
                        s_key[i] = b; s_key[ixj] = a;
                        int t = s_idx[i]; s_idx[i] = s_idx[ixj]; s_idx[ixj] = t;
                    }
                }
            }
            __syncthreads();
        }
    }

    // z = -(s - w),  w[i] = N - i
    for (int i = tid; i < N; i += THREADS)
        s_z[i] = (float)(N - i) - s_key[i];
    __syncthreads();

    // PAVA: non-decreasing isotonic regression of z (serial, O(n), thread 0).
    // iso_dec(y) = -iso_inc(-y) = -z_fitted.
    if (tid == 0) {
        int nb = 0;
        for (int i = 0; i < N; ++i) {
            s_bsum[nb] = s_z[i]; s_bcnt[nb] = 1; ++nb;
            // merge while mean(prev) >= mean(cur)  (cross-multiplied, counts > 0)
            while (nb > 1 &&
                   s_bsum[nb - 2] * (float)s_bcnt[nb - 1] >=
                   s_bsum[nb - 1] * (float)s_bcnt[nb - 2]) {
                s_bsum[nb - 2] += s_bsum[nb - 1];
                s_bcnt[nb - 2] += s_bcnt[nb - 1];
                --nb;
            }
        }
        int pos = 0;
        for (int b = 0; b < nb; ++b) {
            float m = s_bsum[b] / (float)s_bcnt[b];
            for (int c = 0; c < s_bcnt[b]; ++c) s_z[pos++] = m;
        }
    }
    __syncthreads();

    // ret_sorted = s - iso_dec = s + z ; accumulate row sum for the mean
    float part = 0.0f;
    for (int i = tid; i < N; i += THREADS) {
        float r = s_key[i] + s_z[i];
        s_key[i] = r;
        part += r;
    }
    s_red[tid] = part;
    __syncthreads();
    for (int off = THREADS / 2; off > 0; off >>= 1) {
        if (tid < off) s_red[tid] += s_red[tid + off];
        __syncthreads();
    }
    if (tid == 0) s_mean = s_red[0] * (1.0f / (float)N);
    __syncthreads();

    // scatter through inverse permutation, centered: M[row][idx[i]] = ret[i]-mean
    const float mean = s_mean;
    float* mrow = M + row * N;
    for (int i = tid; i < N; i += THREADS)
        mrow[s_idx[i]] = s_key[i] - mean;
}

// ---------------------------------------------------------------------------
// Kernel 2: Gram matrix G = M * M^T (16x16, K=2048) via V_WMMA_F32_16X16X4_F32,
// then Spearman per row pair (r, 8+r) and scalar loss.
//
// f32 A 16x4 layout: lanes 0-15 hold M=lane, VGPR0=K0/VGPR1=K1;
//                    lanes 16-31 hold M=lane-16, VGPR0=K2/VGPR1=K3.
// f32 B 4x16 layout mirrors (row-striped across lanes, K=v / K=v+2 per half),
// so for B = A^T (Gram) both operands use identical per-lane registers.
//
// 8 waves split K; each wave runs 4 independent WMMA accumulator chains
// (unrolled K-slices k0, k0+4, k0+8, k0+12) so loads batch and the XDL pipe
// is not serialized on a single C-operand RAW chain. Partial 16x16
// accumulators are reduced through LDS.
// ---------------------------------------------------------------------------
__global__ __launch_bounds__(THREADS)
void gram_spearman_kernel(const float* __restrict__ M, float* __restrict__ out)
{
    __shared__ float cbuf[8][256];   // [wave][lane*8 + vgpr]
    __shared__ float gbuf[256];      // reduced G entries, index = lane*8 + vgpr

    const int tid  = threadIdx.x;
    const int wv   = tid >> 5;       // wave 0..7
    const int lane = tid & 31;
    const int m    = lane & 15;      // matrix row handled by this lane
    const int koff = (lane >> 4) << 1;  // 0 for lanes 0-15, 2 for lanes 16-31

    const float* base = M + m * N + koff;

    // Warm the near caches with each lane's first-use address (global_prefetch_b8)
    __builtin_prefetch(base + wv * 16, 0, 1);

    v8f c0 = {}, c1 = {}, c2 = {}, c3 = {};
    for (int k0 = wv * 16; k0 < N; k0 += 128) {   // 8 waves x 16 K per iter
        const float* p = base + k0;
        v2f a0 = *(const v2f*)(p + 0);            // K slice k0   .. k0+3
        v2f a1 = *(const v2f*)(p + 4);            // K slice k0+4 .. k0+7
        v2f a2 = *(const v2f*)(p + 8);            // K slice k0+8 .. k0+11
        v2f a3 = *(const v2f*)(p + 12);           // K slice k0+12.. k0+15
        c0 = __builtin_amdgcn_wmma_f32_16x16x4_f32(false, a0, false, a0,
                                                   (short)0, c0, false, false);
        c1 = __builtin_amdgcn_wmma_f32_16x16x4_f32(false, a1, false, a1,
                                                   (short)0, c1, false, false);
        c2 = __builtin_amdgcn_wmma_f32_16x16x4_f32(false, a2, false, a2,
                                                   (short)0, c2, false, false);
        c3 = __builtin_amdgcn_wmma_f32_16x16x4_f32(false, a3, false, a3,
                                                   (short)0, c3, false, false);
    }
    v8f c = (c0 + c1) + (c2 + c3);

#pragma unroll
    for (int v = 0; v < 8; ++v) cbuf[wv][lane * 8 + v] = c[v];
    __syncthreads();

    float g = 0.0f;
#pragma unroll
    for (int w = 0; w < 8; ++w) g += cbuf[w][tid];
    gbuf[tid] = g;
    __syncthreads();

    if (tid == 0) {
        // C/D layout: entry (Mr, Nc): vgpr = Mr % 8, lane = Nc + 16*(Mr >= 8)
        float acc = 0.0f;
        for (int r = 0; r < 8; ++r) {
            float varp = gbuf[r * 8 + r];           // G[r, r]      lane=r,    v=r
            float cov  = gbuf[(8 + r) * 8 + r];     // G[r, 8+r]    lane=8+r,  v=r
            float vart = gbuf[(24 + r) * 8 + r];    // G[8+r, 8+r]  lane=24+r, v=r
            float sp = cov / (sqrtf(varp) * sqrtf(vart) + 1e-8f);
            acc += sp;
        }
        out[0] = 1.0f - acc * 0.125f;
    }
}

// ---------------------------------------------------------------------------
extern "C" void kernel_launch(void* const* d_in, const int* in_sizes, int n_in,
                              void* d_out, int out_size, void* d_ws, size_t ws_size,
                              hipStream_t stream)
{
    const float* pred   = (const float*)d_in[0];   // [8, 2048] f32
    const float* target = (const float*)d_in[1];   // [8, 2048] f32
    float* M = (float*)d_ws;                       // 16*2048 f32 = 128 KB scratch

    softrank_center_kernel<<<16, THREADS, 0, stream>>>(pred, target, M);
    gram_spearman_kernel<<<1, THREADS, 0, stream>>>(M, (float*)d_out);
}